// SimpleVQAutoEncoder_68298569941095
// MI455X (gfx1250) — compile-verified
//
#include <hip/hip_runtime.h>
#include <hip/hip_bf16.h>
#include <math.h>

// ---------------------------------------------------------------------------
// SimpleVQAutoEncoder forward for gfx1250 (MI455X).
// Inputs (f32): x[64,1,256,256], w1[16,1,3,3], b1[16], w2[32,16,3,3], b2[32],
//               codebook[512,32], w3[16,32,3,3], b3[16], w4[1,16,3,3], b4[1]
// Outputs concat (f32): image[64*256*256] | indices[64*4096] | commit_loss[1]
// ---------------------------------------------------------------------------

typedef __attribute__((ext_vector_type(2))) float v2f;
typedef __attribute__((ext_vector_type(8))) float v8f;

#define OUT_IMG   (64u * 256u * 256u)   // 4,194,304
#define IDX_N     (64u * 4096u)         // 262,144
#define LOSS_OFF  (OUT_IMG + IDX_N)     // 4,456,448

__device__ __forceinline__ float gelu_exact(float v) {
    return 0.5f * v * (1.0f + erff(v * 0.70710678118654752440f));
}

// ---------------------------------------------------------------------------
// K1: conv1 (1->16, 3x3 SAME) + maxpool2 + gelu.
// x: [64,256,256] -> h1 channel-last [64, 128*128, 16]
// ---------------------------------------------------------------------------
__global__ void k_conv1_pool_gelu(const float* __restrict__ x,
                                  const float* __restrict__ w1,
                                  const float* __restrict__ b1,
                                  float* __restrict__ h1) {
    __shared__ float ws[16 * 9];
    __shared__ float bs[16];
    int t = threadIdx.x;
    if (t < 144) ws[t] = w1[t];
    if (t < 16)  bs[t] = b1[t];
    __syncthreads();

    int gid = blockIdx.x * blockDim.x + t;          // 64*128*128 threads
    int xq = gid & 127;
    int yq = (gid >> 7) & 127;
    int b  = gid >> 14;
    const float* xp = x + (size_t)b * 65536;

    float p[4][4];
#pragma unroll
    for (int dy = 0; dy < 4; ++dy) {
        int iy = 2 * yq - 1 + dy;
#pragma unroll
        for (int dx = 0; dx < 4; ++dx) {
            int ix = 2 * xq - 1 + dx;
            p[dy][dx] = (iy >= 0 && iy < 256 && ix >= 0 && ix < 256)
                          ? xp[iy * 256 + ix] : 0.0f;
        }
    }

    float* outp = h1 + (size_t)gid * 16;
#pragma unroll
    for (int oc = 0; oc < 16; ++oc) {
        float m = -INFINITY;
#pragma unroll
        for (int py = 0; py < 2; ++py) {
#pragma unroll
            for (int px = 0; px < 2; ++px) {
                float s = bs[oc];
#pragma unroll
                for (int ty = 0; ty < 3; ++ty)
#pragma unroll
                    for (int tx = 0; tx < 3; ++tx)
                        s = fmaf(ws[oc * 9 + ty * 3 + tx], p[py + ty][px + tx], s);
                m = fmaxf(m, s);
            }
        }
        outp[oc] = gelu_exact(m);
    }
}

// ---------------------------------------------------------------------------
// K2: conv2 (16->32, 3x3 SAME on 128-grid) + maxpool2.
// h1 channel-last [64,128*128,16] -> h2 channel-last [64, 64*64, 32] (= x_flat)
// ---------------------------------------------------------------------------
__global__ void k_conv2_pool(const float* __restrict__ h1,
                             const float* __restrict__ w2,
                             const float* __restrict__ b2,
                             float* __restrict__ h2) {
    __shared__ float ws[32 * 16 * 9];   // 18 KB
    __shared__ float bs[32];
    for (int i = threadIdx.x; i < 32 * 16 * 9; i += blockDim.x) ws[i] = w2[i];
    if (threadIdx.x < 32) bs[threadIdx.x] = b2[threadIdx.x];
    __syncthreads();

    int gid = blockIdx.x * blockDim.x + threadIdx.x;  // 64*64*64 threads
    int xq = gid & 63;
    int yq = (gid >> 6) & 63;
    int b  = gid >> 12;
    const float* hp = h1 + (size_t)b * 16384 * 16;

    float mx[32];
#pragma unroll
    for (int oc = 0; oc < 32; ++oc) mx[oc] = -INFINITY;

#pragma unroll
    for (int py = 0; py < 2; ++py) {
#pragma unroll
        for (int px = 0; px < 2; ++px) {
            int cy = 2 * yq + py, cx = 2 * xq + px;
            float acc[32];
#pragma unroll
            for (int oc = 0; oc < 32; ++oc) acc[oc] = bs[oc];
#pragma unroll
            for (int ty = 0; ty < 3; ++ty) {
                int iy = cy + ty - 1;
                if (iy < 0 || iy >= 128) continue;
#pragma unroll
                for (int tx = 0; tx < 3; ++tx) {
                    int ix = cx + tx - 1;
                    if (ix < 0 || ix >= 128) continue;
                    const float* v = hp + ((size_t)iy * 128 + ix) * 16;
                    float vv[16];
#pragma unroll
                    for (int ic = 0; ic < 16; ++ic) vv[ic] = v[ic];
                    int tap = ty * 3 + tx;
#pragma unroll
                    for (int oc = 0; oc < 32; ++oc) {
                        const float* wr = ws + oc * 144 + tap;
#pragma unroll
                        for (int ic = 0; ic < 16; ++ic)
                            acc[oc] = fmaf(wr[ic * 9], vv[ic], acc[oc]);
                    }
                }
            }
#pragma unroll
            for (int oc = 0; oc < 32; ++oc) mx[oc] = fmaxf(mx[oc], acc[oc]);
        }
    }
    float* outp = h2 + (size_t)gid * 32;   // gid == b*4096 + yq*64 + xq
#pragma unroll
    for (int oc = 0; oc < 32; ++oc) outp[oc] = mx[oc];
}

// ---------------------------------------------------------------------------
// K3: VQ via V_WMMA_F32_16X16X4_F32, LDS-staged codebook.
// Block: 256 threads (8 waves). Codebook (64 KB) + c^2 (2 KB) staged in LDS
// once per block; each wave owns 2 row-tiles (32 x-vectors) so every
// ds_load_b64 B-fragment feeds 2 WMMAs while all state stays
// register-resident (~110 VGPRs, no scratch spills).
// dist argmin uses (c2 - 2*xc); x2 is row-constant and cancels in argmin.
// A frag (16x4 f32): lane m=lane&15, VGPR0/1 = K {2*(lane>>4), +1}.
// B frag (4x16 f32): lane n=lane&15, same K striping (codebook row n).
// C/D 16x16 f32: 8 VGPRs; VGPR v holds M = v + 8*(lane>>4), N = lane&15.
// ---------------------------------------------------------------------------
#define VQ_ROWTILES 2   // 16-row tiles per wave

__global__ void k_vq_wmma(const float* __restrict__ X,     // [262144, 32]
                          const float* __restrict__ cb,    // [512, 32]
                          float* __restrict__ qout,        // [262144, 32]
                          float* __restrict__ idx_out,     // [262144] (float)
                          float* __restrict__ loss_out) {  // [1]
    __shared__ float cbs[512 * 32];   // 64 KB codebook, row-major
    __shared__ float c2s[512];        // |c_n|^2

    // Cooperative stage: 4096 float4 across 256 threads.
    for (int i = threadIdx.x; i < 512 * 32 / 4; i += blockDim.x)
        ((float4*)cbs)[i] = ((const float4*)cb)[i];
    __syncthreads();
    for (int i = threadIdx.x; i < 512; i += blockDim.x) {
        const float* cr = cbs + i * 32;
        float s = 0.0f;
#pragma unroll
        for (int k = 0; k < 32; ++k) s = fmaf(cr[k], cr[k], s);
        c2s[i] = s;
    }
    __syncthreads();

    const int lane  = threadIdx.x & 31;
    const int wave  = threadIdx.x >> 5;
    const int wave_base = (blockIdx.x * (blockDim.x >> 5) + wave) * (16 * VQ_ROWTILES);
    const int mrow  = lane & 15;
    const int khalf = lane >> 4;   // 0: K pair {0,1}, 1: K pair {2,3} per step

    // Preload A fragments: VQ_ROWTILES row-tiles x 8 K-steps (32 VGPRs).
    v2f a[VQ_ROWTILES][8];
#pragma unroll
    for (int rt = 0; rt < VQ_ROWTILES; ++rt) {
#pragma unroll
        for (int s = 0; s < 8; ++s) {
            int k = 4 * s + 2 * khalf;
            const float* xr = X + (size_t)(wave_base + rt * 16 + mrow) * 32 + k;
            a[rt][s].x = xr[0];
            a[rt][s].y = xr[1];
        }
    }

    float minval[VQ_ROWTILES][8];
    int   minidx[VQ_ROWTILES][8];
#pragma unroll
    for (int rt = 0; rt < VQ_ROWTILES; ++rt)
#pragma unroll
        for (int v = 0; v < 8; ++v) { minval[rt][v] = INFINITY; minidx[rt][v] = 0; }

    for (int tile = 0; tile < 32; ++tile) {
        int n = tile * 16 + mrow;                 // this lane's code column
        const float* cr = cbs + n * 32;
        float c2 = c2s[n];
        v8f acc[VQ_ROWTILES];
#pragma unroll
        for (int rt = 0; rt < VQ_ROWTILES; ++rt) acc[rt] = (v8f){};
#pragma unroll
        for (int s = 0; s < 8; ++s) {
            int k = 4 * s + 2 * khalf;
            v2f bf;
            bf.x = cr[k];
            bf.y = cr[k + 1];
#pragma unroll
            for (int rt = 0; rt < VQ_ROWTILES; ++rt)
                acc[rt] = __builtin_amdgcn_wmma_f32_16x16x4_f32(
                    false, a[rt][s], false, bf, (short)0, acc[rt], false, false);
        }
#pragma unroll
        for (int rt = 0; rt < VQ_ROWTILES; ++rt) {
#pragma unroll
            for (int v = 0; v < 8; ++v) {
                float score = fmaf(-2.0f, acc[rt][v], c2);
                if (score < minval[rt][v]) { minval[rt][v] = score; minidx[rt][v] = n; }
            }
        }
    }

    // Reduce argmin across the 16 lanes sharing each row (first-min wins ties).
#pragma unroll
    for (int m = 1; m <= 8; m <<= 1) {
#pragma unroll
        for (int rt = 0; rt < VQ_ROWTILES; ++rt) {
#pragma unroll
            for (int v = 0; v < 8; ++v) {
                float ov = __shfl_xor(minval[rt][v], m, 32);
                int   oi = __shfl_xor(minidx[rt][v], m, 32);
                if (ov < minval[rt][v] ||
                    (ov == minval[rt][v] && oi < minidx[rt][v])) {
                    minval[rt][v] = ov;
                    minidx[rt][v] = oi;
                }
            }
        }
    }

    // Emit results per row-tile; gather quantized vectors from LDS codebook.
    float lsum = 0.0f;
#pragma unroll
    for (int rt = 0; rt < VQ_ROWTILES; ++rt) {
        int idxs[16];
#pragma unroll
        for (int v = 0; v < 8; ++v) {
            idxs[v]     = __shfl(minidx[rt][v], 0, 32);   // rows 0..7
            idxs[v + 8] = __shfl(minidx[rt][v], 16, 32);  // rows 8..15
        }
#pragma unroll
        for (int r = 0; r < 16; ++r) {
            int gi   = idxs[r];
            int grow = wave_base + rt * 16 + r;
            float cv = cbs[gi * 32 + lane];
            qout[(size_t)grow * 32 + lane] = cv;
            float d = cv - X[(size_t)grow * 32 + lane];
            lsum = fmaf(d, d, lsum);
            if (lane == r) idx_out[grow] = (float)gi;
        }
    }
#pragma unroll
    for (int m = 1; m <= 16; m <<= 1) lsum += __shfl_xor(lsum, m, 32);
    if (lane == 0)
        atomicAdd(loss_out, lsum * (1.0f / 8388608.0f));  // mean over 64*4096*32
}

// ---------------------------------------------------------------------------
// K4: upsample2 + conv3 (32->16, 3x3 SAME on 128-grid) + gelu.
// q channel-last [64, 64*64, 32] -> h3 channel-last [64, 128*128, 16]
// ---------------------------------------------------------------------------
__global__ void k_up_conv3_gelu(const float* __restrict__ q,
                                const float* __restrict__ w3,
                                const float* __restrict__ b3,
                                float* __restrict__ h3) {
    __shared__ float ws[16 * 32 * 9];   // 18 KB
    __shared__ float bs[16];
    for (int i = threadIdx.x; i < 16 * 32 * 9; i += blockDim.x) ws[i] = w3[i];
    if (threadIdx.x < 16) bs[threadIdx.x] = b3[threadIdx.x];
    __syncthreads();

    int gid = blockIdx.x * blockDim.x + threadIdx.x;  // 64*128*128 threads
    int xq = gid & 127;
    int yq = (gid >> 7) & 127;
    int b  = gid >> 14;
    const float* qp = q + (size_t)b * 4096 * 32;

    float acc[16];
#pragma unroll
    for (int oc = 0; oc < 16; ++oc) acc[oc] = bs[oc];

#pragma unroll
    for (int ty = 0; ty < 3; ++ty) {
        int iy = yq + ty - 1;
        if (iy < 0 || iy >= 128) continue;
#pragma unroll
        for (int tx = 0; tx < 3; ++tx) {
            int ix = xq + tx - 1;
            if (ix < 0 || ix >= 128) continue;
            const float* v = qp + ((size_t)(iy >> 1) * 64 + (ix >> 1)) * 32;
            float vv[32];
#pragma unroll
            for (int ic = 0; ic < 32; ++ic) vv[ic] = v[ic];
            int tap = ty * 3 + tx;
#pragma unroll
            for (int oc = 0; oc < 16; ++oc) {
                const float* wr = ws + oc * 288 + tap;
#pragma unroll
                for (int ic = 0; ic < 32; ++ic)
                    acc[oc] = fmaf(wr[ic * 9], vv[ic], acc[oc]);
            }
        }
    }
    float* outp = h3 + (size_t)gid * 16;
#pragma unroll
    for (int oc = 0; oc < 16; ++oc) outp[oc] = gelu_exact(acc[oc]);
}

// ---------------------------------------------------------------------------
// K5: upsample2 + conv4 (16->1, 3x3 SAME on 256-grid) + clip.
// h3 channel-last [64, 128*128, 16] -> out image [64,256,256]
// ---------------------------------------------------------------------------
__global__ void k_up_conv4_clip(const float* __restrict__ h3,
                                const float* __restrict__ w4,
                                const float* __restrict__ b4,
                                float* __restrict__ out) {
    __shared__ float ws[16 * 9];
    __shared__ float bias;
    int t = threadIdx.x;
    if (t < 144) ws[t] = w4[t];
    if (t == 0) bias = b4[0];
    __syncthreads();

    int gid = blockIdx.x * blockDim.x + t;  // 64*256*256 threads
    int xq = gid & 255;
    int yq = (gid >> 8) & 255;
    int b  = gid >> 16;
    const float* hp = h3 + (size_t)b * 16384 * 16;

    float acc = bias;
#pragma unroll
    for (int ty = 0; ty < 3; ++ty) {
        int iy = yq + ty - 1;
        if (iy < 0 || iy >= 256) continue;
#pragma unroll
        for (int tx = 0; tx < 3; ++tx) {
            int ix = xq + tx - 1;
            if (ix < 0 || ix >= 256) continue;
            const float* v = hp + ((size_t)(iy >> 1) * 128 + (ix >> 1)) * 16;
            int tap = ty * 3 + tx;
#pragma unroll
            for (int ic = 0; ic < 16; ++ic)
                acc = fmaf(ws[ic * 9 + tap], v[ic], acc);
        }
    }
    out[gid] = fminf(1.0f, fmaxf(-1.0f, acc));
}

// ---------------------------------------------------------------------------
extern "C" void kernel_launch(void* const* d_in, const int* in_sizes, int n_in,
                              void* d_out, int out_size, void* d_ws, size_t ws_size,
                              hipStream_t stream) {
    (void)in_sizes; (void)n_in; (void)out_size; (void)ws_size;
    const float* x  = (const float*)d_in[0];
    const float* w1 = (const float*)d_in[1];
    const float* b1 = (const float*)d_in[2];
    const float* w2 = (const float*)d_in[3];
    const float* b2 = (const float*)d_in[4];
    const float* cb = (const float*)d_in[5];
    const float* w3 = (const float*)d_in[6];
    const float* b3 = (const float*)d_in[7];
    const float* w4 = (const float*)d_in[8];
    const float* b4 = (const float*)d_in[9];

    float* out_img  = (float*)d_out;
    float* out_idx  = out_img + OUT_IMG;
    float* out_loss = out_img + LOSS_OFF;

    // Workspace partition (all f32):
    float* h1 = (float*)d_ws;                 // 64*16384*16 = 16,777,216
    float* h2 = h1 + (size_t)64 * 16384 * 16; // 64*4096*32  =  8,388,608
    float* qv = h2 + (size_t)64 * 4096 * 32;  // 64*4096*32  =  8,388,608
    float* h3 = qv + (size_t)64 * 4096 * 32;  // 64*16384*16 = 16,777,216

    hipMemsetAsync(out_loss, 0, sizeof(float), stream);

    k_conv1_pool_gelu<<<4096, 256, 0, stream>>>(x, w1, b1, h1);
    k_conv2_pool<<<1024, 256, 0, stream>>>(h1, w2, b2, h2);
    // 262144 rows / (8 waves * 32 rows) = 1024 blocks of 256 threads
    k_vq_wmma<<<1024, 256, 0, stream>>>(h2, cb, qv, out_idx, out_loss);
    k_up_conv3_gelu<<<4096, 256, 0, stream>>>(qv, w3, b3, h3);
    k_up_conv4_clip<<<16384, 256, 0, stream>>>(h3, w4, b4, out_img);
}